// GraphSAGEModel_26817775796491
// MI455X (gfx1250) — compile-verified
//
#include <hip/hip_runtime.h>
#include <hip/hip_bf16.h>

typedef float v2f_t __attribute__((ext_vector_type(2)));
typedef float v8f_t __attribute__((ext_vector_type(8)));

// 16-byte int vector type matching the async-to-LDS builtin's parameter type.
typedef int v4i_t __attribute__((vector_size(16)));
typedef v4i_t __attribute__((address_space(1))) *gv4i_p;  // global
typedef v4i_t __attribute__((address_space(3))) *lv4i_p;  // LDS

#define D_FEAT 128

// ---------------------------------------------------------------------------
// CDNA5 async global->LDS copy (b128) + ASYNCcnt wait, with safe fallbacks.
// ---------------------------------------------------------------------------
__device__ __forceinline__ void async_cp_b128(const float* g, float* l) {
#if defined(__gfx1250__) && __has_builtin(__builtin_amdgcn_global_load_async_to_lds_b128)
  __builtin_amdgcn_global_load_async_to_lds_b128(
      (gv4i_p)(unsigned long long)g,   // global addrspace(1) v4i*
      (lv4i_p)l,                       // LDS addrspace(3) v4i*
      0, 0);
#else
  *(float4*)l = *(const float4*)g;
#endif
}

__device__ __forceinline__ void wait_async_done() {
#if defined(__gfx1250__) && __has_builtin(__builtin_amdgcn_global_load_async_to_lds_b128)
#if __has_builtin(__builtin_amdgcn_s_wait_asynccnt)
  __builtin_amdgcn_s_wait_asynccnt(0);
#else
  asm volatile("s_wait_asynccnt 0x0" ::: "memory");
#endif
#endif
}

// ---------------------------------------------------------------------------
// Zero-fill (float4 granularity)
// ---------------------------------------------------------------------------
__global__ void fill0_kernel(float4* __restrict__ p, long n4) {
  long i = (long)blockIdx.x * blockDim.x + threadIdx.x;
  if (i < n4) p[i] = make_float4(0.f, 0.f, 0.f, 0.f);
}

// ---------------------------------------------------------------------------
// Edge scatter: one wave32 per edge. Lane l handles feature elements
// [4l, 4l+4): float4 gather from feat[src], 4x global_atomic_add_f32 into
// agg[dst] (L2-resident: 51 MB feature matrix << 192 MB L2). Lane 0 bumps
// the degree counter.
// ---------------------------------------------------------------------------
__global__ __launch_bounds__(256)
void sage_scatter_kernel(const float* __restrict__ feat,
                         const int* __restrict__ src,
                         const int* __restrict__ dst,
                         float* __restrict__ agg,
                         float* __restrict__ cnt,
                         int E) {
  const int lane = threadIdx.x & 31;
  const int edge = blockIdx.x * (blockDim.x >> 5) + (threadIdx.x >> 5);
  if (edge >= E) return;
  const int s = src[edge];
  const int d = dst[edge];
  const float4 v = ((const float4*)(feat + (long)s * D_FEAT))[lane];
  float* dp = agg + (long)d * D_FEAT + lane * 4;
  unsafeAtomicAdd(dp + 0, v.x);
  unsafeAtomicAdd(dp + 1, v.y);
  unsafeAtomicAdd(dp + 2, v.z);
  unsafeAtomicAdd(dp + 3, v.w);
  if (cnt != nullptr && lane == 0) unsafeAtomicAdd(cnt + d, 1.0f);
}

// ---------------------------------------------------------------------------
// Fused SAGE linear: out = (agg * 1/max(cnt,1)) @ Wl^T + x @ Wr^T + b (+ReLU)
//
// Block = 256 threads = 8 waves; each wave owns a 16-row x 128-col output
// tile as 8 v8f accumulators. K runs in four 32-wide panels; both weight
// matrices are staged into a double-buffered 64 KB LDS region with
// global_load_async_to_lds_b128 (panel kb+1 prefetched while panel kb is
// consumed), each wave waiting its own ASYNCcnt before the block barrier.
//
// Mean normalization is folded into the A fragment: per-lane scalar
// inv = 1/max(cnt[row],1) multiplies the two A floats per K step, deleting
// the separate 51 MB normalize pass.
//
// Fragment layouts (ISA 7.12.2):
//   A 16x4 f32: lane L -> M = L&15; K = k0 + 2*(L>>4), K+1
//   B 4x16 f32: lane L -> N = L&15; K = k0 + 2*(L>>4), K+1; B[k][j]=W[j*128+k]
//   C/D 16x16 : VGPR v -> M = v + 8*(L>>4), N = L&15
// ---------------------------------------------------------------------------
__global__ __launch_bounds__(256)
void sage_gemm_kernel(const float* __restrict__ agg,
                      const float* __restrict__ cnt,
                      const float* __restrict__ xin,
                      const float* __restrict__ Wl,
                      const float* __restrict__ Wr,
                      const float* __restrict__ bias,
                      float* __restrict__ out,
                      int N, int do_relu) {
  // [buf][l(0)/r(1)][128 rows x 32-K panel]  = 64 KB
  __shared__ float sW[2][2][D_FEAT * 32];

  const int tid  = threadIdx.x;
  const int lane = tid & 31;
  const int wave = tid >> 5;
  const int n    = lane & 15;   // N (or M for A) within 16-tile
  const int hi   = lane >> 4;   // 0/1 -> K offset 0/2

  const long tile0 = ((long)blockIdx.x * 8 + wave) * 16;
  long arow = tile0 + n;
  if (arow >= N) arow = N - 1;  // clamp: keep EXEC all-1s for WMMA/barriers
  const float* arow_p = agg + arow * D_FEAT;
  const float* xrow_p = xin + arow * D_FEAT;
  const float inv = 1.0f / fmaxf(cnt[arow], 1.0f);  // folded mean scale

  // ---- prologue: async-stage K-panel 0 into buffer 0 ----
#pragma unroll
  for (int i = 0; i < 4; ++i) {
    const int idx = tid + i * 256;        // 0..1023 float4 slots
    const int r   = idx >> 3;
    const int c4  = (idx & 7) << 2;
    async_cp_b128(Wl + r * D_FEAT + c4, &sW[0][0][r * 32 + c4]);
    async_cp_b128(Wr + r * D_FEAT + c4, &sW[0][1][r * 32 + c4]);
  }

  v8f_t acc[8];
#pragma unroll
  for (int jt = 0; jt < 8; ++jt) {
    const float bv = bias[jt * 16 + n];
#pragma unroll
    for (int c = 0; c < 8; ++c) acc[jt][c] = bv;
  }

  wait_async_done();
  __syncthreads();

  for (int kb = 0; kb < 4; ++kb) {
    const int cur = kb & 1;
    // ---- prefetch next panel into the other buffer ----
    if (kb < 3) {
      const int kbase = (kb + 1) * 32;
#pragma unroll
      for (int i = 0; i < 4; ++i) {
        const int idx = tid + i * 256;
        const int r   = idx >> 3;
        const int c4  = (idx & 7) << 2;
        async_cp_b128(Wl + r * D_FEAT + kbase + c4, &sW[cur ^ 1][0][r * 32 + c4]);
        async_cp_b128(Wr + r * D_FEAT + kbase + c4, &sW[cur ^ 1][1][r * 32 + c4]);
      }
    }

    // ---- consume current panel: 8 K-steps x 8 jt x 2 matrices ----
    const int kbase = kb * 32;
#pragma unroll
    for (int kk = 0; kk < 8; ++kk) {
      const int kl = kk * 4 + hi * 2;     // local K in panel
      const int kg = kbase + kl;          // global K
      v2f_t aL = *(const v2f_t*)(arow_p + kg);
      aL.x *= inv; aL.y *= inv;           // mean = agg * 1/max(cnt,1)
      const v2f_t aR = *(const v2f_t*)(xrow_p + kg);
#pragma unroll
      for (int jt = 0; jt < 8; ++jt) {
        const v2f_t bL = *(const v2f_t*)(&sW[cur][0][(jt * 16 + n) * 32 + kl]);
        const v2f_t bR = *(const v2f_t*)(&sW[cur][1][(jt * 16 + n) * 32 + kl]);
        acc[jt] = __builtin_amdgcn_wmma_f32_16x16x4_f32(
            false, aL, false, bL, (short)0, acc[jt], false, false);
        acc[jt] = __builtin_amdgcn_wmma_f32_16x16x4_f32(
            false, aR, false, bR, (short)0, acc[jt], false, false);
      }
    }

    if (kb < 3) {
      wait_async_done();   // our prefetch for panel kb+1 has landed
      __syncthreads();     // everyone's prefetch landed, reads of cur done
    }
  }

  // ---- epilogue: bias pre-folded; optional ReLU; row-predicated stores ----
#pragma unroll
  for (int jt = 0; jt < 8; ++jt) {
#pragma unroll
    for (int v = 0; v < 8; ++v) {
      const long row = tile0 + v + hi * 8;
      if (row < N) {
        float val = acc[jt][v];
        if (do_relu) val = fmaxf(val, 0.0f);
        out[row * D_FEAT + jt * 16 + n] = val;
      }
    }
  }
}

// ---------------------------------------------------------------------------
// Launch: layer1 = scatter+gemm(relu) -> h ; layer2 -> d_out
// ---------------------------------------------------------------------------
extern "C" void kernel_launch(void* const* d_in, const int* in_sizes, int n_in,
                              void* d_out, int out_size, void* d_ws, size_t ws_size,
                              hipStream_t stream) {
  const float* x   = (const float*)d_in[0];
  const int*   ei  = (const int*)d_in[1];
  const float* W1l = (const float*)d_in[2];
  const float* W1r = (const float*)d_in[3];
  const float* b1  = (const float*)d_in[4];
  const float* W2l = (const float*)d_in[5];
  const float* W2r = (const float*)d_in[6];
  const float* b2  = (const float*)d_in[7];
  float* out = (float*)d_out;

  const int N = in_sizes[0] / D_FEAT;
  const int E = in_sizes[1] / 2;
  const int* src = ei;
  const int* dst = ei + E;

  float* agg = (float*)d_ws;                       // N*128 f32
  float* h   = agg + (size_t)N * D_FEAT;           // N*128 f32
  float* cnt = h   + (size_t)N * D_FEAT;           // N f32

  const long nd4 = (long)N * (D_FEAT / 4);         // float4 count of N*128
  const long nc4 = (N + 3) / 4;

  const int zgrid_nd = (int)((nd4 + 255) / 256);
  const int zgrid_nc = (int)((nc4 + 255) / 256);
  const int sgrid    = (E + 7) / 8;                // 8 edges (waves) per block
  const int ggrid    = (N + 127) / 128;            // 8 waves x 16 rows per block

  // ---- Layer 1 ----
  fill0_kernel<<<zgrid_nd, 256, 0, stream>>>((float4*)agg, nd4);
  fill0_kernel<<<zgrid_nc, 256, 0, stream>>>((float4*)cnt, nc4);
  sage_scatter_kernel<<<sgrid, 256, 0, stream>>>(x, src, dst, agg, cnt, E);
  sage_gemm_kernel<<<ggrid, 256, 0, stream>>>(agg, cnt, x, W1l, W1r, b1, h, N, 1);

  // ---- Layer 2 ----
  fill0_kernel<<<zgrid_nd, 256, 0, stream>>>((float4*)agg, nd4);
  sage_scatter_kernel<<<sgrid, 256, 0, stream>>>(h, src, dst, agg, nullptr, E);
  sage_gemm_kernel<<<ggrid, 256, 0, stream>>>(agg, cnt, h, W2l, W2r, b2, out, N, 0);
}